// LinOSSLayer_67723044323807
// MI455X (gfx1250) — compile-verified
//
#include <hip/hip_runtime.h>
#include <hip/hip_bf16.h>

// LinOSS-IM layer for MI455X (gfx1250, wave32).
// Shapes fixed by the harness setup: B=4, L=8192, H=128, P=64, N=B*L=32768.
// Pipeline (all on `stream`, in order):
//   K0 pack:     build W1 (H x 2P) = [B_real;B_imag]^T, W2 (2P x H) = [C_real; -C_imag],
//                and the per-channel 2x2 recurrence coefficients M11..M22, F-scales.
//   K1 gemm<0>:  Bu = u @ W1           (fp32 WMMA 16x16x4, 32768x128 @ 128x128)
//   K2 scanA:    local chunk scans (T=128, NC=64), in-place x-part over Bu
//   K3 scanB:    M^T + carry prefix over the 64 chunks (512 threads)
//   K4 scanC:    fixup: X[t] += (M^{t+1} * carry).x
//   K5 gemm<1>:  out = X @ W2 + u * D  (fp32 WMMA + fused residual)

typedef __attribute__((ext_vector_type(2))) float v2f;
typedef __attribute__((ext_vector_type(4))) float v4f;
typedef __attribute__((ext_vector_type(8))) float v8f;

namespace {
constexpr int Bb  = 4;
constexpr int Ll  = 8192;
constexpr int Hh  = 128;      // == 2P
constexpr int Pp  = 64;
constexpr int Nn  = Bb * Ll;  // 32768 rows
constexpr int Tt  = 128;      // chunk length
constexpr int NC  = Ll / Tt;  // 64 chunks
}

// ---------------------------------------------------------------- K0: pack weights + coefficients
__global__ __launch_bounds__(128) void pack_coef_kernel(
    const float* __restrict__ A_diag, const float* __restrict__ B_real,
    const float* __restrict__ B_imag, const float* __restrict__ C_real,
    const float* __restrict__ C_imag, const float* __restrict__ steps,
    float* __restrict__ w1, float* __restrict__ w2, float* __restrict__ coef)
{
  const int h = blockIdx.x;   // 0..127
  const int j = threadIdx.x;  // 0..127  (j = p + 64*c, c=0 real / c=1 imag)
  // W1[h][j]: input projection, K = H rows, N = 2P cols
  w1[h * 128 + j] = (j < 64) ? B_real[j * Hh + h] : B_imag[(j - 64) * Hh + h];
  // W2[j][h]: output projection, K = 2P rows, N = H cols (minus sign on imag baked in)
  w2[j * 128 + h] = (j < 64) ? C_real[h * Pp + j] : -C_imag[h * Pp + (j - 64)];
  if (h == 0 && j < Pp) {
    const float A  = fmaxf(A_diag[j], 0.0f);          // relu
    const float s  = 1.0f / (1.0f + expf(-steps[j])); // sigmoid
    const float sc = 1.0f / fmaf(s * s, A, 1.0f);     // schur
    const float m11 = 1.0f - s * s * A * sc;
    const float m12 = -s * A * sc;
    const float m21 = s * sc;
    const float m22 = sc;
    coef[0 * 64 + j] = m11;
    coef[1 * 64 + j] = m12;
    coef[2 * 64 + j] = m21;
    coef[3 * 64 + j] = m22;
    coef[4 * 64 + j] = m11 * s;  // F1 scale
    coef[5 * 64 + j] = m21 * s;  // F2 scale
  }
}

// ---------------------------------------------------------------- K1/K5: fp32 WMMA GEMM, N x 128 @ 128 x 128
// One wave32 per 16x16 output tile; 8 waves (256 threads) cover all 8 column
// tiles of one 16-row strip. W (128x128 = 64 KB) staged in LDS per block.
// Fragment layouts per CDNA5 ISA 7.12.2 (32-bit, 16x16x4):
//   A: lane = {n = M row (lane&15), half = lane>>4}; VGPR0 = A[n][k0], VGPR1 = A[n][k0+1], k0 = kk + 2*half
//   B: lane = {n = N col};                           VGPR0 = B[k0][n], VGPR1 = B[k0+1][n]
//   C: VGPR i = C[i + 8*half][n]
template <int MODE>  // 0: plain store, 1: += u*D residual
__global__ __launch_bounds__(256) void gemm128_wmma_kernel(
    const float* __restrict__ Xin, const float* __restrict__ W,
    float* __restrict__ Y, const float* __restrict__ u, const float* __restrict__ Dv)
{
  __shared__ float sW[128 * 128];  // 64 KB
  const int tid = threadIdx.x;
  {
    const v4f* Wv = (const v4f*)W;
    v4f* sWv = (v4f*)sW;
#pragma unroll
    for (int i = 0; i < 16; ++i) sWv[tid + 256 * i] = Wv[tid + 256 * i];
  }
  __syncthreads();

  const int rowTile = blockIdx.x;         // 0..N/16-1
  const int wave    = tid >> 5;           // 0..7 -> column tile
  const int lane    = tid & 31;
  const int n       = lane & 15;
  const int half    = lane >> 4;          // 0/1
  const int colBase = wave * 16;
  const int row     = rowTile * 16 + n;   // this lane's A row

  const v2f* arow = (const v2f*)(Xin + (size_t)row * 128);
  v8f c = {};
#pragma unroll
  for (int kk = 0; kk < 128; kk += 4) {
    const int k0 = kk + 2 * half;
    v2f a = arow[k0 >> 1];                             // A[row][k0], A[row][k0+1]
    v2f b;
    b.x = sW[k0 * 128 + colBase + n];                  // B[k0][col]
    b.y = sW[(k0 + 1) * 128 + colBase + n];            // B[k0+1][col]
    c = __builtin_amdgcn_wmma_f32_16x16x4_f32(
        /*neg_a=*/false, a, /*neg_b=*/false, b,
        /*c_mod=*/(short)0, c, /*reuse_a=*/false, /*reuse_b=*/false);
  }

  const size_t rbase = (size_t)rowTile * 16 * 128;
  const int col = colBase + n;
  const float d = (MODE == 1) ? Dv[col] : 0.0f;
#pragma unroll
  for (int i = 0; i < 8; ++i) {
    const size_t idx = rbase + (size_t)(i + 8 * half) * 128 + col;
    float v = c[i];
    if (MODE == 1) v = fmaf(u[idx], d, v);
    Y[idx] = v;
  }
}

// ---------------------------------------------------------------- K2: local chunk scans (zero init), in place
__global__ __launch_bounds__(128) void scan_local_kernel(
    float* __restrict__ X, const float* __restrict__ coef,
    float* __restrict__ cfz, float* __restrict__ cfx)
{
  const int j  = threadIdx.x;          // 0..127 = p + 64*c
  const int p  = j & 63;
  const int b  = blockIdx.x >> 6;      // NC = 64
  const int ch = blockIdx.x & 63;
  const float m11 = coef[p], m12 = coef[64 + p];
  const float m21 = coef[128 + p], m22 = coef[192 + p];
  const float f1s = coef[256 + p], f2s = coef[320 + p];
  size_t base = ((size_t)(b * Ll + ch * Tt)) * 128 + j;
  float z = 0.0f, x = 0.0f;
#pragma unroll 4
  for (int t = 0; t < Tt; ++t) {
    const float f  = X[base + (size_t)t * 128];
    const float zn = fmaf(m11, z, fmaf(m12, x, f1s * f));
    const float xn = fmaf(m21, z, fmaf(m22, x, f2s * f));
    z = zn; x = xn;
    X[base + (size_t)t * 128] = x;     // keep position part only
  }
  const int idx = blockIdx.x * 128 + j;
  cfz[idx] = z; cfx[idx] = x;
}

// ---------------------------------------------------------------- K3: M^T + carry prefix over chunks
__global__ __launch_bounds__(128) void scan_combine_kernel(
    const float* __restrict__ coef, const float* __restrict__ cfz,
    const float* __restrict__ cfx, float* __restrict__ czv, float* __restrict__ cxv)
{
  const int j = threadIdx.x, p = j & 63, b = blockIdx.x;
  const float m11 = coef[p], m12 = coef[64 + p];
  const float m21 = coef[128 + p], m22 = coef[192 + p];
  // MT = M^T via T homogeneous steps on the identity columns
  float c1z = 1.0f, c1x = 0.0f, c2z = 0.0f, c2x = 1.0f;
  for (int t = 0; t < Tt; ++t) {
    const float a = fmaf(m11, c1z, m12 * c1x), bb = fmaf(m21, c1z, m22 * c1x);
    const float d = fmaf(m11, c2z, m12 * c2x), e  = fmaf(m21, c2z, m22 * c2x);
    c1z = a; c1x = bb; c2z = d; c2x = e;
  }
  float az = 0.0f, ax = 0.0f;  // exclusive carry for chunk 0 is zero
  for (int ch = 0; ch < NC; ++ch) {
    const int idx = (b * NC + ch) * 128 + j;
    czv[idx] = az; cxv[idx] = ax;
    const float nz = fmaf(c1z, az, fmaf(c2z, ax, cfz[idx]));
    const float nx = fmaf(c1x, az, fmaf(c2x, ax, cfx[idx]));
    az = nz; ax = nx;
  }
}

// ---------------------------------------------------------------- K4: fixup X[t] += (M^{t+1} * carry).x
__global__ __launch_bounds__(128) void scan_fixup_kernel(
    float* __restrict__ X, const float* __restrict__ coef,
    const float* __restrict__ czv, const float* __restrict__ cxv)
{
  const int j  = threadIdx.x, p = j & 63;
  const int b  = blockIdx.x >> 6;
  const int ch = blockIdx.x & 63;
  if (ch == 0) return;  // carry is exactly zero
  const float m11 = coef[p], m12 = coef[64 + p];
  const float m21 = coef[128 + p], m22 = coef[192 + p];
  const int idx = blockIdx.x * 128 + j;
  float hz = czv[idx], hx = cxv[idx];
  size_t base = ((size_t)(b * Ll + ch * Tt)) * 128 + j;
#pragma unroll 4
  for (int t = 0; t < Tt; ++t) {
    const float nz = fmaf(m11, hz, m12 * hx);
    const float nx = fmaf(m21, hz, m22 * hx);
    hz = nz; hx = nx;
    X[base + (size_t)t * 128] += hx;
  }
}

// ---------------------------------------------------------------- launch
extern "C" void kernel_launch(void* const* d_in, const int* in_sizes, int n_in,
                              void* d_out, int out_size, void* d_ws, size_t ws_size,
                              hipStream_t stream) {
  const float* u      = (const float*)d_in[0];  // (B, L, H)
  const float* A_diag = (const float*)d_in[1];  // (P,)
  const float* B_real = (const float*)d_in[2];  // (P, H)
  const float* B_imag = (const float*)d_in[3];  // (P, H)
  const float* C_real = (const float*)d_in[4];  // (H, P)
  const float* C_imag = (const float*)d_in[5];  // (H, P)
  const float* Dv     = (const float*)d_in[6];  // (H,)
  const float* steps  = (const float*)d_in[7];  // (P,)
  float* out = (float*)d_out;

  // Workspace layout (floats)
  float* w1   = (float*)d_ws;                  // 16384
  float* w2   = w1 + 16384;                    // 16384
  float* coef = w2 + 16384;                    // 384
  float* xbuf = coef + 384;                    // N*128 (Bu, then x in place)
  float* cfz  = xbuf + (size_t)Nn * 128;       // B*NC*128
  float* cfx  = cfz + Bb * NC * 128;
  float* czv  = cfx + Bb * NC * 128;
  float* cxv  = czv + Bb * NC * 128;
  const size_t need = (size_t)(cxv + Bb * NC * 128 - (float*)d_ws) * sizeof(float);
  if (ws_size < need) return;  // ~17.4 MB required

  pack_coef_kernel<<<128, 128, 0, stream>>>(A_diag, B_real, B_imag, C_real, C_imag,
                                            steps, w1, w2, coef);
  gemm128_wmma_kernel<0><<<Nn / 16, 256, 0, stream>>>(u, w1, xbuf, nullptr, nullptr);
  scan_local_kernel<<<Bb * NC, 128, 0, stream>>>(xbuf, coef, cfz, cfx);
  scan_combine_kernel<<<Bb, 128, 0, stream>>>(coef, cfz, cfx, czv, cxv);
  scan_fixup_kernel<<<Bb * NC, 128, 0, stream>>>(xbuf, coef, czv, cxv);
  gemm128_wmma_kernel<1><<<Nn / 16, 256, 0, stream>>>(xbuf, w2, out, u, Dv);
}